// PeriodicPrimitives2D_25022479467069
// MI455X (gfx1250) — compile-verified
//
#include <hip/hip_runtime.h>

// CDNA5 / gfx1250, wave32.
typedef __attribute__((ext_vector_type(2))) float v2f;
typedef __attribute__((ext_vector_type(8))) float v8f;

#define WPB   8        // waves per block (8 * 32 = 256 threads)
#define LDS_Q 2048     // float4 slots of LDS staging (32 KB)

// MAX_FREQUENCY / NUM_FREQUENCIES; stored frequency is in *revolutions* so
// __builtin_amdgcn_cosf (v_cos_f32: cos(2*pi*x)) needs no 1/(2pi) multiply.
#define FREQ_SCALE (1024.0f / 1024.0f)

template <int KT>
__global__ __launch_bounds__(WPB * 32)
void gabor_splat_kernel(const float* __restrict__ x,
                        const float* __restrict__ colors,
                        const float* __restrict__ positions,
                        const float* __restrict__ scales,
                        const float* __restrict__ rotations,
                        const float* __restrict__ coeffs,
                        const int*   __restrict__ indices,
                        float* __restrict__ out,
                        int N, int G, int Krt, int gchunk)
{
    const int K  = (KT > 0) ? KT : Krt;   // top-k frequencies per dim
    const int S4 = 3 + K;                 // float4 slots per gaussian record
    __shared__ float4 smem[LDS_Q];

    const int tid  = threadIdx.x;
    const int lane = tid & 31;
    const int warp = tid >> 5;
    const int half = lane >> 4;           // 0: lanes 0-15, 1: lanes 16-31
    const int prow = lane & 15;           // point row (A) / output column (B,C)
    const int tile_base = (blockIdx.x * WPB + warp) * 16;

    // Each lane owns one query point of the wave's 16-point tile.
    int p  = tile_base + prow;
    int pc = (p < N) ? p : (N > 0 ? N - 1 : 0);
    const float xx = x[2 * pc + 0];
    const float xy = x[2 * pc + 1];

    // Loop-invariant per-lane channel-select masks (branch-free B build).
    const float m0 = (prow == 0) ? 1.0f : 0.0f;
    const float m1 = (prow == 1) ? 1.0f : 0.0f;
    const float m2 = (prow == 2) ? 1.0f : 0.0f;

    v8f acc = {0.f, 0.f, 0.f, 0.f, 0.f, 0.f, 0.f, 0.f};  // C/D 16x16 f32

    for (int g0 = 0; g0 < G; g0 += gchunk) {
        int count = G - g0; if (count > gchunk) count = gchunk;
        int count4 = (count + 3) & ~3;    // pad to WMMA K-group of 4

        __syncthreads();                  // protect previous chunk's readers
        // Cooperative stage: one gaussian per thread. Record (float4 slots):
        //  [0] {px, py, cos(rot), sin(rot)}
        //  [1] {sx, sy, c0, c1}
        //  [2] {c2, 0, 0, 0}
        //  [3+k] {fx_k (rev), fy_k (rev), cx_k, cy_k}
        for (int gl = tid; gl < count4; gl += blockDim.x) {
            float4* rec = &smem[gl * S4];
            int gi = g0 + gl;
            if (gl < count) {
                float r = rotations[gi];
                float4 q0 = { positions[2*gi+0], positions[2*gi+1],
                              __cosf(r), __sinf(r) };
                float4 q1 = { scales[2*gi+0], scales[2*gi+1],
                              colors[3*gi+0], colors[3*gi+1] };
                float4 q2 = { colors[3*gi+2], 0.f, 0.f, 0.f };
                rec[0] = q0; rec[1] = q1; rec[2] = q2;
                for (int k = 0; k < K; ++k) {
                    int base = (gi * K + k) * 2;
                    float4 qk = { (float)indices[base+0] * FREQ_SCALE,
                                  (float)indices[base+1] * FREQ_SCALE,
                                  coeffs[base+0], coeffs[base+1] };
                    rec[3 + k] = qk;
                }
            } else {
                // Zero record -> finite A value, zero color -> zero contribution.
                float4 z = {0.f, 0.f, 0.f, 0.f};
                for (int q = 0; q < S4; ++q) rec[q] = z;
            }
        }
        __syncthreads();

        // 4 gaussians per WMMA step. ISA A layout (16x4 f32): lane M = prow,
        // VGPR j holds K-slice j + 2*half -> this lane computes gaussians
        // gb + 2*half + j for j = 0,1. B (4x16) mirrors the same K rows.
        // Running pointers: no per-iteration index multiply.
        const float4* rec0 = &smem[(2 * half + 0) * S4];
        const float4* rec1 = &smem[(2 * half + 1) * S4];
        const int step = 4 * S4;
        for (int gb = 0; gb < count4; gb += 4, rec0 += step, rec1 += step) {
            float av[2], bv[2];
            #pragma unroll
            for (int j = 0; j < 2; ++j) {
                const float4* rec = (j == 0) ? rec0 : rec1;
                float4 p0 = rec[0];                 // ds_load_b128 broadcast
                float dx = xx - p0.x, dy = xy - p0.y;
                float tx = dx * p0.z + dy * p0.w;   // rotate into local frame
                float ty = dy * p0.z - dx * p0.w;
                float4 p1 = rec[1];
                float ax = tx * p1.x, ay = ty * p1.y;
                float gv = __expf(-0.5f * (ax * ax + ay * ay));  // v_exp_f32
                float wx = 0.f, wy = 0.f;
                #pragma unroll
                for (int k = 0; k < K; ++k) {
                    float4 pk = rec[3 + k];
                    wx += pk.z * __builtin_amdgcn_cosf(pk.x * tx); // v_cos_f32
                    wy += pk.w * __builtin_amdgcn_cosf(pk.y * ty);
                }
                av[j] = gv * wx * wy;
                // Branch-free B column build: colors padded to 16 columns.
                float c2v = rec[2].x;
                bv[j] = p1.z * m0 + p1.w * m1 + c2v * m2;
            }
            v2f A = { av[0], av[1] };
            v2f B = { bv[0], bv[1] };
            // D[m][n] += sum_{k<4} gval(point m, gauss gb+k) * color[gb+k][n]
            acc = __builtin_amdgcn_wmma_f32_16x16x4_f32(
                      false, A, false, B, (short)0, acc, false, false);
        }
    }

    // C/D layout: VGPR v, half h -> row M = v + 8*h, column N = prow.
    if (prow < 3) {
        #pragma unroll
        for (int v = 0; v < 8; ++v) {
            int row = tile_base + v + 8 * half;
            if (row < N) out[row * 3 + prow] = acc[v];
        }
    }
}

extern "C" void kernel_launch(void* const* d_in, const int* in_sizes, int n_in,
                              void* d_out, int out_size, void* d_ws, size_t ws_size,
                              hipStream_t stream) {
    const float* x         = (const float*)d_in[0];
    const float* colors    = (const float*)d_in[1];
    const float* positions = (const float*)d_in[2];
    const float* scales    = (const float*)d_in[3];
    const float* rotations = (const float*)d_in[4];
    const float* coeffs    = (const float*)d_in[5];
    const int*   indices   = (const int*)d_in[6];
    float* out = (float*)d_out;

    const int N = in_sizes[0] / 2;           // x: [N,2]
    const int G = in_sizes[1] / 3;           // colors: [G,3]
    const int K = (G > 0) ? in_sizes[5] / (2 * G) : 1;  // coeffs: [G,K,2]

    const int S4 = 3 + K;
    int gchunk = (LDS_Q / S4) & ~3;          // gaussians per LDS chunk (mult of 4)
    if (gchunk < 4) gchunk = 4;

    const int tiles  = (N + 15) / 16;
    const int blocks = (tiles + WPB - 1) / WPB;
    if (blocks <= 0) return;

    if (K == 4) {
        gabor_splat_kernel<4><<<blocks, WPB * 32, 0, stream>>>(
            x, colors, positions, scales, rotations, coeffs, indices, out,
            N, G, K, gchunk);
    } else {
        gabor_splat_kernel<0><<<blocks, WPB * 32, 0, stream>>>(
            x, colors, positions, scales, rotations, coeffs, indices, out,
            N, G, K, gchunk);
    }
}